// GraphSage_53635551592820
// MI455X (gfx1250) — compile-verified
//
#include <hip/hip_runtime.h>
#include <hip/hip_bf16.h>
#include <math.h>

#define N_NODES 100000
#define N_EDGES 1600000
#define F_IN    64            // feature width entering every layer's aggregation
#define KCAT    128           // [mean | h] concatenated K

typedef __attribute__((ext_vector_type(2))) float v2f;
typedef __attribute__((ext_vector_type(8))) float v8f;

// ---------------------------------------------------------------- utilities
__global__ void gs_fill_zero(float* __restrict__ p, int n) {
    int i = blockIdx.x * blockDim.x + threadIdx.x;
    if (i < n) p[i] = 0.0f;
}

__global__ void gs_deg_count(const int* __restrict__ dst, float* __restrict__ deg) {
    int e = blockIdx.x * blockDim.x + threadIdx.x;
    if (e < N_EDGES) atomicAdd(&deg[dst[e]], 1.0f);
}

__global__ void gs_deg_inv(float* __restrict__ deg) {
    int i = blockIdx.x * blockDim.x + threadIdx.x;
    if (i < N_NODES) {
        float d = deg[i];
        deg[i] = 1.0f / fmaxf(d, 1.0f);
    }
}

// Edge scatter: 16 threads per edge, each moves a float4 chunk of the 64-wide row.
// h[src] gather is coalesced (256B per 16 lanes); scatter is 4x global_atomic_add_f32.
__global__ void gs_scatter(const float* __restrict__ h,
                           const int* __restrict__ src,
                           const int* __restrict__ dst,
                           float* __restrict__ agg) {
    long long tid = (long long)blockIdx.x * blockDim.x + threadIdx.x;
    if (tid >= (long long)N_EDGES * 16) return;
    int e = (int)(tid >> 4);
    int c = (int)(tid & 15);
    int s = src[e];
    int d = dst[e];
    float4 v = ((const float4*)(h + (size_t)s * F_IN))[c];
    float* p = agg + (size_t)d * F_IN + c * 4;
    atomicAdd(p + 0, v.x);
    atomicAdd(p + 1, v.y);
    atomicAdd(p + 2, v.z);
    atomicAdd(p + 3, v.w);
}

// ------------------------------------------------------------- fused GEMM
// out[16 nodes, Nout] = ([agg*deg_inv | h] (16x128)) @ ([Wl ; Wr] (128xNout)) + b
// One block = 16 nodes; wave w owns N-tile [16*w, 16*w+16). blockDim = 32*(Nout/16).
// K=128 swept in 32 steps of V_WMMA_F32_16X16X4_F32.
__global__ void gs_sage_gemm(const float* __restrict__ agg,
                             const float* __restrict__ deg_inv,
                             const float* __restrict__ hin,
                             const float* __restrict__ Wl,
                             const float* __restrict__ Wr,
                             const float* __restrict__ bias,
                             float* __restrict__ out,
                             int Nout, int applyElu) {
    __shared__ float As[16 * 132];   // 16 rows x 128 cols, stride 132 (bank-safe)
    __shared__ float Ws[128 * 68];   // 128 rows x <=64 cols, stride Nout+4

    const int tid     = threadIdx.x;
    const int nthr    = blockDim.x;
    const int node0   = blockIdx.x * 16;
    const int wstride = Nout + 4;

    // Stage A = [mean | h] with deg_inv folded in. 2048 scalar elems, coalesced in k.
    for (int i = tid; i < 16 * KCAT; i += nthr) {
        int r = i >> 7;           // node row 0..15
        int k = i & 127;          // 0..127
        int node = node0 + r;
        float v;
        if (k < F_IN) {
            v = agg[(size_t)node * F_IN + k] * deg_inv[node];
        } else {
            v = hin[(size_t)node * F_IN + (k - F_IN)];
        }
        As[r * 132 + k] = v;
    }
    // Stage B = [Wl ; Wr] stacked along K.
    for (int i = tid; i < KCAT * Nout; i += nthr) {
        int k = i / Nout;
        int n = i - k * Nout;
        float v = (k < F_IN) ? Wl[k * Nout + n] : Wr[(k - F_IN) * Nout + n];
        Ws[k * wstride + n] = v;
    }
    __syncthreads();

    const int lane = tid & 31;
    const int wave = tid >> 5;
    const int n0   = wave * 16;
    const int lm   = lane & 15;   // A-row for this lane == B-col for this lane
    const int hi   = lane >> 4;   // lane half selects K sub-pair per ISA layout

    v8f acc = {};
#pragma unroll
    for (int k0 = 0; k0 < KCAT; k0 += 4) {
        v2f a, b;
        // A 16x4 f32 layout: vgpr0 = K {0|2}, vgpr1 = K {1|3} by lane half
        a.x = As[lm * 132 + k0 + 2 * hi];
        a.y = As[lm * 132 + k0 + 2 * hi + 1];
        // B 4x16 f32 layout: row striped across lanes, same K split by lane half
        b.x = Ws[(k0 + 2 * hi) * wstride + n0 + lm];
        b.y = Ws[(k0 + 2 * hi + 1) * wstride + n0 + lm];
        acc = __builtin_amdgcn_wmma_f32_16x16x4_f32(
            /*neg_a=*/false, a, /*neg_b=*/false, b,
            /*c_mod=*/(short)0, acc, /*reuse_a=*/false, /*reuse_b=*/false);
    }

    // Epilogue: bias + optional ELU, store 16x16 tile.
    float bv = bias[n0 + lm];
#pragma unroll
    for (int i = 0; i < 8; i++) {
        float v = acc[i] + bv;
        if (applyElu) v = (v > 0.0f) ? v : (__expf(v) - 1.0f);
        int m = i + 8 * hi;       // C/D layout: lanes 0-15 -> M=vgpr, 16-31 -> M=vgpr+8
        out[(size_t)(node0 + m) * Nout + n0 + lm] = v;
    }
}

// ---------------------------------------------------------------- driver
extern "C" void kernel_launch(void* const* d_in, const int* in_sizes, int n_in,
                              void* d_out, int out_size, void* d_ws, size_t ws_size,
                              hipStream_t stream) {
    const float* x   = (const float*)d_in[0];
    const int*   ei  = (const int*)d_in[1];
    const float* Wl0 = (const float*)d_in[2];
    const float* Wr0 = (const float*)d_in[3];
    const float* b0  = (const float*)d_in[4];
    const float* Wl1 = (const float*)d_in[5];
    const float* Wr1 = (const float*)d_in[6];
    const float* b1  = (const float*)d_in[7];
    const float* Wl2 = (const float*)d_in[8];
    const float* Wr2 = (const float*)d_in[9];
    const float* b2  = (const float*)d_in[10];

    const int* src = ei;             // edge_index[0]
    const int* dst = ei + N_EDGES;   // edge_index[1]

    float* ws  = (float*)d_ws;
    float* deg = ws;                          // N_NODES (holds deg_inv after gs_deg_inv)
    float* agg = ws + 100096;                 // N_NODES * 64
    float* h1  = agg + (size_t)N_NODES * F_IN;
    float* h2  = h1  + (size_t)N_NODES * F_IN;
    float* outp = (float*)d_out;              // N_NODES * 32

    const int T = 256;
    const int aggElems = N_NODES * F_IN;
    dim3 gFillDeg((N_NODES + T - 1) / T);
    dim3 gFillAgg((aggElems + T - 1) / T);
    dim3 gEdges((N_EDGES + T - 1) / T);
    dim3 gScatter((int)(((long long)N_EDGES * 16 + T - 1) / T));
    dim3 gGemm(N_NODES / 16);                 // 6250, exact

    // degree -> deg_inv (shared by all layers)
    gs_fill_zero<<<gFillDeg, T, 0, stream>>>(deg, N_NODES);
    gs_deg_count<<<gEdges, T, 0, stream>>>(dst, deg);
    gs_deg_inv<<<gFillDeg, T, 0, stream>>>(deg);

    // layer 0: x -> h1 (ELU)
    gs_fill_zero<<<gFillAgg, T, 0, stream>>>(agg, aggElems);
    gs_scatter<<<gScatter, T, 0, stream>>>(x, src, dst, agg);
    gs_sage_gemm<<<gGemm, 128, 0, stream>>>(agg, deg, x, Wl0, Wr0, b0, h1, 64, 1);

    // layer 1: h1 -> h2 (ELU)
    gs_fill_zero<<<gFillAgg, T, 0, stream>>>(agg, aggElems);
    gs_scatter<<<gScatter, T, 0, stream>>>(h1, src, dst, agg);
    gs_sage_gemm<<<gGemm, 128, 0, stream>>>(agg, deg, h1, Wl1, Wr1, b1, h2, 64, 1);

    // layer 2: h2 -> out (no ELU, Nout=32)
    gs_fill_zero<<<gFillAgg, T, 0, stream>>>(agg, aggElems);
    gs_scatter<<<gScatter, T, 0, stream>>>(h2, src, dst, agg);
    gs_sage_gemm<<<gGemm, 64, 0, stream>>>(agg, deg, h2, Wl2, Wr2, b2, outp, 32, 0);
}